// DLModel_63256278335700
// MI455X (gfx1250) — compile-verified
//
#include <hip/hip_runtime.h>
#include <hip/hip_bf16.h>
#include <math.h>

typedef float v2f __attribute__((ext_vector_type(2)));
typedef float v8f __attribute__((ext_vector_type(8)));

// Fast activations: single v_exp_f32 + single v_rcp_f32 (~1 ulp), no IEEE div
// expansion. The recurrences are contracting, so the ~1e-7 per-step error does
// not accumulate meaningfully over 4096 steps.
__device__ __forceinline__ float sigm(float x) {
    return __builtin_amdgcn_rcpf(1.0f + __expf(-x));
}
__device__ __forceinline__ float tanh_f(float x) {
    float r = __builtin_amdgcn_rcpf(1.0f + __expf(2.0f * x));
    return fmaf(-2.0f, r, 1.0f);
}

// One scalar LSTM cell with hidden size 1 (gate order i,f,g,o).
__device__ __forceinline__ void cell1(float xt, float& h, float& c,
                                      const float wi[4], const float wh[4],
                                      const float bb[4]) {
    float zi = bb[0] + wi[0] * xt + wh[0] * h;
    float zf = bb[1] + wi[1] * xt + wh[1] * h;
    float zg = bb[2] + wi[2] * xt + wh[2] * h;
    float zo = bb[3] + wi[3] * xt + wh[3] * h;
    float i = sigm(zi), f = sigm(zf), g = tanh_f(zg), o = sigm(zo);
    c = f * c + i * g;
    h = o * tanh_f(c);
}

// Kernel A:
//  - blocks [0, gridDim-2]: noise LSTM seq pass + autoregressive pass
//    (one batch element per lane; scalar recurrences, H=1)
//  - block  gridDim-1: main LSTM (batch 0 only, H=4) via
//    V_WMMA_F32_16X16X4_F32, fused with fc + lv recurrence -> lvs[] in ws
__global__ __launch_bounds__(32) void k_recurrences(
    const float* __restrict__ x, const float* __restrict__ ts,
    const float* __restrict__ pre_lv,
    const float* __restrict__ l0Wih, const float* __restrict__ l0Whh,
    const float* __restrict__ l0bih, const float* __restrict__ l0bhh,
    const float* __restrict__ l1Wih, const float* __restrict__ l1Whh,
    const float* __restrict__ l1bih, const float* __restrict__ l1bhh,
    const float* __restrict__ n0Wih, const float* __restrict__ n0Whh,
    const float* __restrict__ n0bih, const float* __restrict__ n0bhh,
    const float* __restrict__ n1Wih, const float* __restrict__ n1Whh,
    const float* __restrict__ n1bih, const float* __restrict__ n1bhh,
    const float* __restrict__ fcW, const float* __restrict__ fcb,
    const float* __restrict__ nfcW, const float* __restrict__ nfcb,
    float* __restrict__ out, float* __restrict__ lvs,
    int B, int S) {
    if ((int)blockIdx.x == (int)gridDim.x - 1) {
        // ---------------- main LSTM (batch 0) + lv recurrence ----------------
        const int l  = threadIdx.x;       // lane 0..31, wave32
        const int n  = l & 15;            // gate column this lane owns
        const int j  = l & 3;             // hidden unit this lane owns
        const int kb = (l >> 4) << 1;     // K-pair index for A/B-matrix layout

        const float wih0 = l0Wih[n];
        const float b0   = l0bih[n] + l0bhh[n];
        const float b1   = l1bih[n] + l1bhh[n];
        // B (4x16, KxN) layout: lane = N + 16*(K>=2), vgpr = K&1 ; B(k,n)=W[n,k]
        v2f Bh0, Bi1, Bh1;
        Bh0.x = l0Whh[n * 4 + kb]; Bh0.y = l0Whh[n * 4 + kb + 1];
        Bi1.x = l1Wih[n * 4 + kb]; Bi1.y = l1Wih[n * 4 + kb + 1];
        Bh1.x = l1Whh[n * 4 + kb]; Bh1.y = l1Whh[n * 4 + kb + 1];

        const float fw00 = fcW[0], fw01 = fcW[1], fw02 = fcW[2], fw03 = fcW[3];
        const float fw10 = fcW[4], fw11 = fcW[5], fw12 = fcW[6], fw13 = fcW[7];
        const float fb0 = fcb[0], fb1 = fcb[1];

        // Recurrent A operands (A(M,K): every row = (h0..h3)); reused between
        // layer-1 input this step and layer-0 recurrence next step.
        v2f Ah0; Ah0.x = 0.f; Ah0.y = 0.f;
        v2f Ah1; Ah1.x = 0.f; Ah1.y = 0.f;
        float c0o = 0.f, c1o = 0.f;       // per-lane cell state (hidden unit j)
        float lv = pre_lv[0];
        int t = 0;
        const float LVC = 11313.0f * 0.5f / (1250.0f * 230.0f);

        for (int s = 0; s < S; ++s) {
            const float xt  = x[s];        // x[0, s, 0]
            const float tss = ts[s];

            // ---- layer 0: z = x*Wih0 + h0 @ Whh0^T + b (WMMA, K=4) ----
            v8f C0 = {};
            C0[0] = b0 + xt * wih0;        // rows 0 and 8 of C carry bias+x term
            v8f D0 = __builtin_amdgcn_wmma_f32_16x16x4_f32(
                false, Ah0, false, Bh0, (short)0, C0, false, false);
            float z  = D0[0];              // z[n] in lane n (D rows 0 and 8 identical)
            float sg = sigm(z), th = tanh_f(z);
            float si = __shfl(sg,      j, 32);
            float sf = __shfl(sg,  4 + j, 32);
            float tg = __shfl(th,  8 + j, 32);
            float so = __shfl(sg, 12 + j, 32);
            c0o = sf * c0o + si * tg;
            float h0o = so * tanh_f(c0o);
            Ah0.x = __shfl(h0o, kb,     32);   // rebuild A from new h0
            Ah0.y = __shfl(h0o, kb + 1, 32);

            // ---- layer 1: z = h0 @ Wih1^T + h1 @ Whh1^T + b (2 chained WMMAs) ----
            v8f C1 = {};
            C1[0] = b1;
            v8f D1 = __builtin_amdgcn_wmma_f32_16x16x4_f32(
                false, Ah0, false, Bi1, (short)0, C1, false, false);
            v8f D2 = __builtin_amdgcn_wmma_f32_16x16x4_f32(
                false, Ah1, false, Bh1, (short)0, D1, false, false);
            z  = D2[0];
            sg = sigm(z); th = tanh_f(z);
            si = __shfl(sg,      j, 32);
            sf = __shfl(sg,  4 + j, 32);
            tg = __shfl(th,  8 + j, 32);
            so = __shfl(sg, 12 + j, 32);
            c1o = sf * c1o + si * tg;
            float h1o = so * tanh_f(c1o);
            Ah1.x = __shfl(h1o, kb,     32);
            Ah1.y = __shfl(h1o, kb + 1, 32);

            // ---- fc (row 0 of params) + lv physics steps (side chain) ----
            float h1a = __shfl(h1o, 0, 32), h1b = __shfl(h1o, 1, 32);
            float h1c = __shfl(h1o, 2, 32), h1d = __shfl(h1o, 3, 32);
            float p0 = fb0 + h1a * fw00 + h1b * fw01 + h1c * fw02 + h1d * fw03;
            float p1 = fb1 + h1a * fw10 + h1b * fw11 + h1c * fw12 + h1d * fw13;
            int nrep = (int)(tss * 2.0f);  // int(ts/0.5)
            for (int k = 0; k < nrep; ++k) {
                float H3 = (lv < 633.0f) ? 0.0f : (lv - 633.0f);
                float Ht = p0 + 1300.0f - H3;
                lv += sqrtf(19.6f * Ht) * p1 * LVC;
                if (l == 0 && t < S) lvs[t] = lv;
                ++t;
            }
        }
    } else {
        // ---------------- noise LSTM: seq pass + AR pass (H=1) ----------------
        int b = blockIdx.x * 32 + threadIdx.x;
        if (b >= B) return;
        float wi0[4], wh0[4], bb0[4], wi1[4], wh1[4], bb1[4];
#pragma unroll
        for (int g = 0; g < 4; ++g) {
            wi0[g] = n0Wih[g]; wh0[g] = n0Whh[g]; bb0[g] = n0bih[g] + n0bhh[g];
            wi1[g] = n1Wih[g]; wh1[g] = n1Whh[g]; bb1[g] = n1bih[g] + n1bhh[g];
        }
        float h0 = 0.f, c0 = 0.f, h1 = 0.f, c1 = 0.f;
        const float* xb   = x + (size_t)b * S;
        float* nout = out + 2ull * (size_t)B * S + (size_t)b * S;  // noise_out
        float* fin  = out + (size_t)b * S;                         // stash noiseOutput

        float xt = xb[0];
        for (int s = 0; s < S; ++s) {
            float xnext = (s + 1 < S) ? xb[s + 1] : 0.f;   // hide load latency
            cell1(xt, h0, c0, wi0, wh0, bb0);
            cell1(h0, h1, c1, wi1, wh1, bb1);
            nout[s] = h1;
            xt = xnext;
        }
        const float w = nfcW[0], bias = nfcb[0];
        for (int s = 0; s < S; ++s) {
            float xin = h1;                 // feed back last-layer hidden
            cell1(xin, h0, c0, wi0, wh0, bb0);
            cell1(h0,  h1, c1, wi1, wh1, bb1);
            fin[s] = w * h1 + bias;         // noiseOutput; foward added in pass 2
        }
    }
}

// Kernel B: final = noiseOutput + foward ; also materialize foward_out.
// foward_out[b,t] = lvs[(b*S + t) / B]  (raw reshape of broadcast lvs)
__global__ __launch_bounds__(256) void k_combine(const float* __restrict__ lvs,
                                                 float* __restrict__ out,
                                                 int B, int S) {
    size_t N = (size_t)B * S;
    size_t i = (size_t)blockIdx.x * blockDim.x + threadIdx.x;
    if (i >= N) return;
    float f = lvs[i / (size_t)B];
    out[i]     = out[i] + f;   // finalOutput
    out[N + i] = f;            // foward_out
}

extern "C" void kernel_launch(void* const* d_in, const int* in_sizes, int n_in,
                              void* d_out, int out_size, void* d_ws, size_t ws_size,
                              hipStream_t stream) {
    const float* x   = (const float*)d_in[0];
    const float* ts  = (const float*)d_in[1];
    // d_in[2] = phs: only its (nonzero) length matters -> reshape branch is fixed
    const float* pre = (const float*)d_in[3];
    const float* l0Wih = (const float*)d_in[4];
    const float* l0Whh = (const float*)d_in[5];
    const float* l0bih = (const float*)d_in[6];
    const float* l0bhh = (const float*)d_in[7];
    const float* l1Wih = (const float*)d_in[8];
    const float* l1Whh = (const float*)d_in[9];
    const float* l1bih = (const float*)d_in[10];
    const float* l1bhh = (const float*)d_in[11];
    const float* n0Wih = (const float*)d_in[12];
    const float* n0Whh = (const float*)d_in[13];
    const float* n0bih = (const float*)d_in[14];
    const float* n0bhh = (const float*)d_in[15];
    const float* n1Wih = (const float*)d_in[16];
    const float* n1Whh = (const float*)d_in[17];
    const float* n1bih = (const float*)d_in[18];
    const float* n1bhh = (const float*)d_in[19];
    const float* fcW   = (const float*)d_in[20];
    const float* fcb   = (const float*)d_in[21];
    const float* nfcW  = (const float*)d_in[22];
    const float* nfcb  = (const float*)d_in[23];

    const int S = in_sizes[1];              // 4096
    const int B = in_sizes[0] / S;          // 512
    float* out = (float*)d_out;
    float* lvs = (float*)d_ws;              // S floats of scratch

    const int nbNoise = (B + 31) / 32;      // 16 noise blocks (1 wave each)
    k_recurrences<<<dim3(nbNoise + 1), dim3(32), 0, stream>>>(
        x, ts, pre,
        l0Wih, l0Whh, l0bih, l0bhh, l1Wih, l1Whh, l1bih, l1bhh,
        n0Wih, n0Whh, n0bih, n0bhh, n1Wih, n1Whh, n1bih, n1bhh,
        fcW, fcb, nfcW, nfcb, out, lvs, B, S);

    const size_t N = (size_t)B * S;
    const unsigned nblk = (unsigned)((N + 255) / 256);
    k_combine<<<dim3(nblk), dim3(256), 0, stream>>>(lvs, out, B, S);
}